// BackBone_20074677141869
// MI455X (gfx1250) — compile-verified
//
#include <hip/hip_runtime.h>
#include <hip/hip_bf16.h>

// ---- problem constants (from the reference) ----
#define T_  24
#define NA_ 128
#define ND_ 6000
#define E_  20000
#define L_  8000
#define H_  128
#define FF_ 512
#define NH_ 8

typedef __attribute__((ext_vector_type(2))) float v2f;
typedef __attribute__((ext_vector_type(8))) float v8f;

static __device__ __forceinline__ v8f wmma4(v2f a, v2f b, v8f c) {
  // V_WMMA_F32_16X16X4_F32: D = A(16x4) * B(4x16) + C, full fp32
  return __builtin_amdgcn_wmma_f32_16x16x4_f32(false, a, false, b, (short)0, c,
                                               false, false);
}

static __device__ __forceinline__ v8f vzero8() {
  v8f z = {0.f,0.f,0.f,0.f,0.f,0.f,0.f,0.f};
  return z;
}

static __device__ __forceinline__ float wave_sum(float v) {
#pragma unroll
  for (int m = 16; m >= 1; m >>= 1) v += __shfl_xor(v, m, 32);
  return v;
}

// =====================================================================
// Generic GEMM: C[M,128] = A[M,K] @ W[K,128] (+bias[n]) (+addMat[gidx[m],n])
// block = 256 thr (8 waves); wave -> 16 rows x 64 cols; grid.x in {0,1} cols
// =====================================================================
__global__ void k_gemm(const float* __restrict__ A, const float* __restrict__ W,
                       const float* __restrict__ bias,
                       const float* __restrict__ addMat, const int* __restrict__ gidx,
                       float* __restrict__ C, int M, int K) {
  const int lane = threadIdx.x & 31, wave = threadIdx.x >> 5;
  const int mt = blockIdx.y * 8 + wave;
  if (mt * 16 >= M) return;
  const int row0 = mt * 16, n0 = blockIdx.x * 64;
  const int mloc = lane & 15, ko = (lane >> 4) * 2;
  const int nloc = lane & 15, half = lane >> 4;
  const float* arow = A + (long long)(row0 + mloc) * K + ko;
  v8f acc[4];
#pragma unroll
  for (int j = 0; j < 4; ++j) acc[j] = vzero8();
  for (int kk = 0; kk < K; kk += 4) {
    v2f a = *(const v2f*)(arow + kk);
    const float* wp = W + (kk + ko) * H_ + n0 + nloc;
#pragma unroll
    for (int j = 0; j < 4; ++j) {
      v2f b; b.x = wp[j * 16]; b.y = wp[j * 16 + H_];
      acc[j] = wmma4(a, b, acc[j]);
    }
  }
#pragma unroll
  for (int j = 0; j < 4; ++j) {
    const int n = n0 + j * 16 + nloc;
    const float bv = bias ? bias[n] : 0.f;
#pragma unroll
    for (int r = 0; r < 8; ++r) {
      const int m = row0 + r + half * 8;
      float v = acc[j][r] + bv;
      if (addMat) v += addMat[(long long)gidx[m] * H_ + n];
      C[(long long)m * H_ + n] = v;
    }
  }
}

// =====================================================================
// GNN update: C[z] = relu(A1[z]@W1 + A2[z]@W2), all [*,128]@[128,128]
// =====================================================================
__global__ void k_gemm2(const float* __restrict__ A1, long long sA1,
                        const float* __restrict__ W1g,
                        const float* __restrict__ A2, long long sA2,
                        const float* __restrict__ W2g,
                        float* __restrict__ C, long long sC, int M) {
  const int lane = threadIdx.x & 31, wave = threadIdx.x >> 5;
  const int mt = blockIdx.y * 8 + wave;
  if (mt * 16 >= M) return;
  const int z = blockIdx.z;
  A1 += (long long)z * sA1; A2 += (long long)z * sA2; C += (long long)z * sC;
  const int row0 = mt * 16, n0 = blockIdx.x * 64;
  const int mloc = lane & 15, ko = (lane >> 4) * 2;
  const int nloc = lane & 15, half = lane >> 4;
  const float* a1r = A1 + (long long)(row0 + mloc) * H_ + ko;
  const float* a2r = A2 + (long long)(row0 + mloc) * H_ + ko;
  v8f acc[4];
#pragma unroll
  for (int j = 0; j < 4; ++j) acc[j] = vzero8();
  for (int kk = 0; kk < H_; kk += 4) {
    v2f a1 = *(const v2f*)(a1r + kk);
    v2f a2 = *(const v2f*)(a2r + kk);
    const float* w1p = W1g + (kk + ko) * H_ + n0 + nloc;
    const float* w2p = W2g + (kk + ko) * H_ + n0 + nloc;
#pragma unroll
    for (int j = 0; j < 4; ++j) {
      v2f b1; b1.x = w1p[j * 16]; b1.y = w1p[j * 16 + H_];
      v2f b2; b2.x = w2p[j * 16]; b2.y = w2p[j * 16 + H_];
      acc[j] = wmma4(a1, b1, acc[j]);
      acc[j] = wmma4(a2, b2, acc[j]);
    }
  }
#pragma unroll
  for (int j = 0; j < 4; ++j)
#pragma unroll
    for (int r = 0; r < 8; ++r) {
      const int m = row0 + r + half * 8;
      const int n = n0 + j * 16 + nloc;
      const float v = acc[j][r];
      C[(long long)m * H_ + n] = v > 0.f ? v : 0.f;
    }
}

// =====================================================================
// Edge messages: per edge compute e = attr@W_e + b_e (DE=16), scatter
//   m_d[t,dst] += relu(ha[src]+e);  m_a[t,src] += relu(hd[dst]+e)
// one wave per edge; 4 channels per lane; f32 global atomics
// =====================================================================
__global__ void k_edge(const float* __restrict__ eattr, const int* __restrict__ esrc,
                       const int* __restrict__ edst, const float* __restrict__ We,
                       const float* __restrict__ be,
                       const float* __restrict__ ha, long long sHa,
                       const float* __restrict__ hd, long long sHd,
                       float* __restrict__ md, float* __restrict__ ma) {
  const int lane = threadIdx.x & 31, wave = threadIdx.x >> 5;
  const int t = blockIdx.y;
  const int i = blockIdx.x * 8 + wave;
  if (i >= E_) return;
  const int src = esrc[t * E_ + i];
  const int dst = edst[t * E_ + i];
  const float* ap = eattr + ((long long)t * E_ + i) * 16;
  const float av = (lane < 16) ? ap[lane] : 0.f;
  const int h = lane * 4;
  float e[4] = {be[h], be[h + 1], be[h + 2], be[h + 3]};
#pragma unroll
  for (int d = 0; d < 16; ++d) {
    const float a = __shfl(av, d, 32);
    const float* wp = We + d * H_ + h;
    e[0] += a * wp[0]; e[1] += a * wp[1]; e[2] += a * wp[2]; e[3] += a * wp[3];
  }
  const float* hs  = ha + (long long)t * sHa + (long long)src * H_ + h;
  const float* hdd = hd + (long long)t * sHd + (long long)dst * H_ + h;
  float* mdp = md + ((long long)t * ND_ + dst) * H_ + h;
  float* map = ma + ((long long)t * NA_ + src) * H_ + h;
#pragma unroll
  for (int c = 0; c < 4; ++c) {
    atomicAdd(&mdp[c], fmaxf(hs[c] + e[c], 0.f));
    atomicAdd(&map[c], fmaxf(hdd[c] + e[c], 0.f));
  }
}

// =====================================================================
// Sinusoidal positional encoding add: X[t,n,h] += pe[t,h]
// =====================================================================
__global__ void k_posenc(float* __restrict__ X, int N) {
  const long long total = (long long)T_ * N * H_;
  const long long p = ((long long)blockIdx.x * blockDim.x + threadIdx.x) * 4;
  if (p >= total) return;
  const int t = (int)(p / ((long long)N * H_));
  const int h = (int)(p & (H_ - 1));
#pragma unroll
  for (int c = 0; c < 4; ++c) {
    const int hc = h + c;
    const float freq = expf(-logf(10000.f) * (float)(hc & ~1) / (float)H_);
    const float ang = (float)t * freq;
    X[p + c] += (hc & 1) ? cosf(ang) : sinf(ang);
  }
}

// =====================================================================
// Fused causal MHA core: per (8-column tile, head) workgroup.
// Stage 1: Q/K/V = X@W via WMMA into LDS. Stage 2: T=24 causal softmax
// + A@V per column on VALU. Writes pre-Wo output slice to O.
// =====================================================================
__global__ void k_mha(const float* __restrict__ X, const float* __restrict__ Wq,
                      const float* __restrict__ Wk, const float* __restrict__ Wv,
                      float* __restrict__ O, int N) {
  __shared__ float QKV[3][192 * 16];   // rows r = t*8 + nl, 36 KB
  const int lane = threadIdx.x & 31, wave = threadIdx.x >> 5;
  const int n0 = blockIdx.x * 8;
  const int h0 = blockIdx.y * 16;
  const int mloc = lane & 15, ko = (lane >> 4) * 2;
  const int nloc = lane & 15, half = lane >> 4;
  // ---- stage 1: 12 M-tiles x 3 matrices = 36 WMMA tile jobs ----
  for (int job = wave; job < 36; job += 8) {
    const int mat = job / 12, mt = job % 12;
    const float* Wp = (mat == 0) ? Wq : (mat == 1) ? Wk : Wv;
    const int r = mt * 16 + mloc;          // row in [0,192)
    const int tt = r >> 3, nl = r & 7;
    const float* arow = X + ((long long)tt * N + n0 + nl) * H_ + ko;
    v8f acc = vzero8();
    for (int kk = 0; kk < H_; kk += 4) {
      v2f a = *(const v2f*)(arow + kk);
      const float* wp = Wp + (kk + ko) * H_ + h0 + nloc;
      v2f b; b.x = wp[0]; b.y = wp[H_];
      acc = wmma4(a, b, acc);
    }
    float* out = QKV[mat];
#pragma unroll
    for (int r8 = 0; r8 < 8; ++r8) {
      const int m = mt * 16 + r8 + half * 8;
      out[m * 16 + nloc] = acc[r8];
    }
  }
  __syncthreads();
  // ---- stage 2: wave w owns column n=w, lane = timestep t ----
  const int n = wave;
  const int t = lane;
  if (t < T_) {
    float q[16];
#pragma unroll
    for (int d = 0; d < 16; ++d) q[d] = QKV[0][(t * 8 + n) * 16 + d];
    float sc[T_];
    float mx = -1e30f;
#pragma unroll
    for (int s = 0; s < T_; ++s) {
      float a = 0.f;
#pragma unroll
      for (int d = 0; d < 16; ++d) a += q[d] * QKV[1][(s * 8 + n) * 16 + d];
      a *= 0.25f;                      // 1/sqrt(hd=16)
      sc[s] = (s <= t) ? a : -1e9f;
      mx = fmaxf(mx, sc[s]);
    }
    float sum = 0.f;
#pragma unroll
    for (int s = 0; s < T_; ++s) { sc[s] = expf(sc[s] - mx); sum += sc[s]; }
    const float inv = 1.f / sum;
    float o[16];
#pragma unroll
    for (int d = 0; d < 16; ++d) o[d] = 0.f;
#pragma unroll
    for (int s = 0; s < T_; ++s) {
      const float a = sc[s] * inv;
#pragma unroll
      for (int d = 0; d < 16; ++d) o[d] += a * QKV[2][(s * 8 + n) * 16 + d];
    }
    float* op = O + ((long long)t * N + n0 + n) * H_ + h0;
#pragma unroll
    for (int d = 0; d < 16; ++d) op[d] = o[d];
  }
}

// =====================================================================
// Fused FFN: Y = relu(X@W1[128,512]) @ W2[512,128]; 32-row blocks,
// hidden staged in 64 KB LDS (avoids a 294 MB global intermediate)
// =====================================================================
__global__ void k_ff(const float* __restrict__ X, const float* __restrict__ W1,
                     const float* __restrict__ W2, float* __restrict__ Y, int M) {
  __shared__ float mid[32 * FF_];      // 64 KB
  const int lane = threadIdx.x & 31, wave = threadIdx.x >> 5;
  const int row0 = blockIdx.x * 32;
  const int mloc = lane & 15, ko = (lane >> 4) * 2;
  const int nloc = lane & 15, half = lane >> 4;
  const int mt = wave & 1, jg = wave >> 1;   // 2 m-tiles, 4 col groups
  // ---- stage 1: mid = relu(X@W1), wave covers 16 rows x 128 cols ----
  {
    const float* arow = X + (long long)(row0 + mt * 16 + mloc) * H_ + ko;
    v8f acc[8];
#pragma unroll
    for (int j = 0; j < 8; ++j) acc[j] = vzero8();
    for (int kk = 0; kk < H_; kk += 4) {
      v2f a = *(const v2f*)(arow + kk);
      const float* wp = W1 + (kk + ko) * FF_ + jg * 128 + nloc;
#pragma unroll
      for (int j = 0; j < 8; ++j) {
        v2f b; b.x = wp[j * 16]; b.y = wp[j * 16 + FF_];
        acc[j] = wmma4(a, b, acc[j]);
      }
    }
#pragma unroll
    for (int j = 0; j < 8; ++j) {
      const int nn = jg * 128 + j * 16 + nloc;
#pragma unroll
      for (int r = 0; r < 8; ++r) {
        const int m = mt * 16 + r + half * 8;
        const float v = acc[j][r];
        mid[m * FF_ + nn] = v > 0.f ? v : 0.f;
      }
    }
  }
  __syncthreads();
  // ---- stage 2: Y = mid @ W2, wave covers 16 rows x 32 cols ----
  {
    const float* arow = &mid[(mt * 16 + mloc) * FF_ + ko];
    v8f acc[2];
    acc[0] = vzero8(); acc[1] = vzero8();
    for (int kk = 0; kk < FF_; kk += 4) {
      v2f a = *(const v2f*)(arow + kk);
      const float* wp = W2 + (kk + ko) * H_ + jg * 32 + nloc;
#pragma unroll
      for (int j = 0; j < 2; ++j) {
        v2f b; b.x = wp[j * 16]; b.y = wp[j * 16 + H_];
        acc[j] = wmma4(a, b, acc[j]);
      }
    }
#pragma unroll
    for (int j = 0; j < 2; ++j)
#pragma unroll
      for (int r = 0; r < 8; ++r) {
        const int m = row0 + mt * 16 + r + half * 8;
        const int nn = jg * 32 + j * 16 + nloc;
        Y[(long long)m * H_ + nn] = acc[j][r];
      }
  }
}

// =====================================================================
// X[p,:] = ln(X[p,:] + R[rr,:]) * g + b ; rr = rmod ? p%rmod : p
// one wave per 128-wide row; two-pass mean/var via lane shuffles
// =====================================================================
__global__ void k_res_ln(float* __restrict__ X, const float* __restrict__ R,
                         const float* __restrict__ g, const float* __restrict__ b,
                         int rows, int rmod) {
  const int lane = threadIdx.x & 31, wave = threadIdx.x >> 5;
  const int p = blockIdx.x * 8 + wave;
  if (p >= rows) return;
  const int rr = rmod ? (p % rmod) : p;
  float* xp = X + (long long)p * H_;
  const float* rp = R + (long long)rr * H_;
  const int h = lane * 4;
  float v[4];
#pragma unroll
  for (int c = 0; c < 4; ++c) v[c] = xp[h + c] + rp[h + c];
  float s = v[0] + v[1] + v[2] + v[3];
  s = wave_sum(s);
  const float mean = s * (1.f / H_);
  float ss = 0.f;
#pragma unroll
  for (int c = 0; c < 4; ++c) { const float d = v[c] - mean; ss += d * d; }
  ss = wave_sum(ss);
  const float inv = rsqrtf(ss * (1.f / H_) + 1e-5f);
#pragma unroll
  for (int c = 0; c < 4; ++c)
    xp[h + c] = (v[c] - mean) * inv * g[h + c] + b[h + c];
}

// =====================================================================
// Link prediction: out[t,l] = sum_h dA[t,ls]*dD[t,ld]*w_lp ; wave/pair
// =====================================================================
__global__ void k_link(const float* __restrict__ DA, const float* __restrict__ DD,
                       const int* __restrict__ ls, const int* __restrict__ ld,
                       const float* __restrict__ wlp, float* __restrict__ out,
                       int total) {
  const int lane = threadIdx.x & 31, wave = threadIdx.x >> 5;
  const int p = blockIdx.x * 8 + wave;
  if (p >= total) return;
  const int t = p / L_;
  const int u = ls[p], v = ld[p];
  const int h = lane * 4;
  const float* up = DA + ((long long)t * NA_ + u) * H_ + h;
  const float* vp = DD + ((long long)t * ND_ + v) * H_ + h;
  float s = up[0] * vp[0] * wlp[h] + up[1] * vp[1] * wlp[h + 1] +
            up[2] * vp[2] * wlp[h + 2] + up[3] * vp[3] * wlp[h + 3];
  s = wave_sum(s);
  if (lane == 0) out[p] = s;
}

// =====================================================================
// host orchestration
// =====================================================================
struct DecP {
  const float *W1, *W2, *Wk, *Wo, *Woc, *Wq, *Wv, *Wvc;
  const float *b1, *b2, *b3, *g1, *g2, *g3;
};

extern "C" void kernel_launch(void* const* d_in, const int* in_sizes, int n_in,
                              void* d_out, int out_size, void* d_ws, size_t ws_size,
                              hipStream_t stream) {
  (void)in_sizes; (void)n_in; (void)out_size; (void)ws_size;
  // JAX pytree (sorted-key) flatten order of setup_inputs()
  const int*   drug_ids   = (const int*)  d_in[0];
  const float* edge_attr  = (const float*)d_in[1];
  const int*   edge_dst   = (const int*)  d_in[2];
  const int*   edge_src   = (const int*)  d_in[3];
  const int*   labels_dst = (const int*)  d_in[4];
  const int*   labels_src = (const int*)  d_in[5];
  const float* W_adm  = (const float*)d_in[6];
  const float* W_drug = (const float*)d_in[7];
  const float* W_e    = (const float*)d_in[8];
  const float* b_adm  = (const float*)d_in[9];
  const float* b_drug = (const float*)d_in[10];
  const float* b_e    = (const float*)d_in[11];
  auto F = [&](int i) { return (const float*)d_in[i]; };
  const DecP da = {F(12), F(13), F(14), F(15), F(16), F(17), F(18), F(19),
                   F(20), F(21), F(22), F(23), F(24), F(25)};
  const DecP dd = {F(26), F(27), F(28), F(29), F(30), F(31), F(32), F(33),
                   F(34), F(35), F(36), F(37), F(38), F(39)};
  const float* emb  = F(40);
  const float* Wma  = F(41);
  const float* Wmd  = F(42);
  const float* Wsa  = F(43);
  const float* Wsd  = F(44);
  const float* w_lp = F(45);
  const float* x_adm  = F(46);
  const float* x_drug = F(47);

  // ---- workspace bump allocator ----
  char* wsp = (char*)d_ws;
  auto alloc = [&](size_t bytes) -> float* {
    float* p = (float*)wsp;
    wsp += (bytes + 255) & ~(size_t)255;
    return p;
  };
  const size_t SA = (size_t)T_ * NA_ * H_ * 4;   // 1.5 MB
  const size_t SD = (size_t)T_ * ND_ * H_ * 4;   // 73.7 MB
  float* ha0 = alloc((size_t)NA_ * H_ * 4);
  float* hd0 = alloc((size_t)ND_ * H_ * 4);
  float* xa  = alloc(SA);
  float* haL = alloc(SA);
  float* m_a = alloc(SA);
  float* oa  = alloc(SA);
  float* ya  = alloc(SA);
  float* ca1 = alloc((size_t)NA_ * H_ * 4);
  float* ca2 = alloc((size_t)NA_ * H_ * 4);
  float* cd1 = alloc((size_t)ND_ * H_ * 4);
  float* cd2 = alloc((size_t)ND_ * H_ * 4);
  float* bufD1 = alloc(SD);
  float* bufD2 = alloc(SD);
  float* bufD3 = alloc(SD);

  const dim3 blk(256);
  const int HH = H_ * H_;

  // ---- input projections ----
  k_gemm<<<dim3(2, 1),  blk, 0, stream>>>(x_adm,  W_adm,  b_adm,  nullptr, nullptr, ha0, NA_, 64);
  k_gemm<<<dim3(2, 47), blk, 0, stream>>>(x_drug, W_drug, b_drug, emb, drug_ids,   hd0, ND_, 64);

  // ---- GNN (2 layers, all 24 timesteps in parallel via grid.z) ----
  for (int l = 0; l < 2; ++l) {
    hipMemsetAsync(m_a,   0, SA, stream);
    hipMemsetAsync(bufD3, 0, SD, stream);
    const float* haCur = (l == 0) ? ha0 : haL;
    const long long sHa = (l == 0) ? 0 : (long long)NA_ * H_;
    const float* hdCur = (l == 0) ? hd0 : bufD1;
    const long long sHd = (l == 0) ? 0 : (long long)ND_ * H_;
    k_edge<<<dim3(E_ / 8, T_), blk, 0, stream>>>(edge_attr, edge_src, edge_dst,
                                                 W_e, b_e, haCur, sHa, hdCur, sHd,
                                                 bufD3, m_a);
    float* haOut = (l == 0) ? haL : xa;
    float* hdOut = (l == 0) ? bufD1 : bufD2;
    k_gemm2<<<dim3(2, 1, T_),  blk, 0, stream>>>(haCur, sHa, Wsa + l * HH,
                                                 m_a, (long long)NA_ * H_, Wma + l * HH,
                                                 haOut, (long long)NA_ * H_, NA_);
    k_gemm2<<<dim3(2, 47, T_), blk, 0, stream>>>(hdCur, sHd, Wsd + l * HH,
                                                 bufD3, (long long)ND_ * H_, Wmd + l * HH,
                                                 hdOut, (long long)ND_ * H_, ND_);
  }

  // ---- add positional encoding ----
  k_posenc<<<dim3((T_ * NA_ * H_ / 4 + 255) / 256), blk, 0, stream>>>(xa, NA_);
  k_posenc<<<dim3((T_ * ND_ * H_ / 4 + 255) / 256), blk, 0, stream>>>(bufD2, ND_);

  // ---- transformer decoder (post-norm, 2 layers) ----
  auto decode = [&](float* X, int N, const float* h0, float* obuf, float* ybuf,
                    float* c1, float* c2, const DecP& dp) {
    const int rows = T_ * N;
    const int gy  = (rows / 16 + 7) / 8;
    const int gyc = (N / 16 + 7) / 8;
    for (int l = 0; l < 2; ++l) {
      // self-attention
      k_mha<<<dim3(N / 8, NH_), blk, 0, stream>>>(X, dp.Wq + l * HH, dp.Wk + l * HH,
                                                  dp.Wv + l * HH, obuf, N);
      k_gemm<<<dim3(2, gy), blk, 0, stream>>>(obuf, dp.Wo + l * HH, nullptr,
                                              nullptr, nullptr, ybuf, rows, H_);
      k_res_ln<<<dim3((rows + 7) / 8), blk, 0, stream>>>(X, ybuf, dp.g1 + l * H_,
                                                         dp.b1 + l * H_, rows, 0);
      // cross-attention to length-1 memory == (h0@Wvc)@Woc broadcast over t
      k_gemm<<<dim3(2, gyc), blk, 0, stream>>>(h0, dp.Wvc + l * HH, nullptr,
                                               nullptr, nullptr, c1, N, H_);
      k_gemm<<<dim3(2, gyc), blk, 0, stream>>>(c1, dp.Woc + l * HH, nullptr,
                                               nullptr, nullptr, c2, N, H_);
      k_res_ln<<<dim3((rows + 7) / 8), blk, 0, stream>>>(X, c2, dp.g2 + l * H_,
                                                         dp.b2 + l * H_, rows, N);
      // feed-forward (fused through LDS)
      k_ff<<<dim3(rows / 32), blk, 0, stream>>>(X, dp.W1 + l * H_ * FF_,
                                                dp.W2 + l * FF_ * H_, ybuf, rows);
      k_res_ln<<<dim3((rows + 7) / 8), blk, 0, stream>>>(X, ybuf, dp.g3 + l * H_,
                                                         dp.b3 + l * H_, rows, 0);
    }
  };
  decode(xa,    NA_, ha0, oa,    ya,    ca1, ca2, da);
  decode(bufD2, ND_, hd0, bufD1, bufD3, cd1, cd2, dd);

  // ---- link prediction ----
  k_link<<<dim3((T_ * L_) / 8), blk, 0, stream>>>(xa, bufD2, labels_src, labels_dst,
                                                  w_lp, (float*)d_out, T_ * L_);
}